// EventMamba_80281528697065
// MI455X (gfx1250) — compile-verified
//
#include <hip/hip_runtime.h>
#include <cmath>

#define CDIV(a,b) (((a)+(b)-1)/(b))

typedef __attribute__((ext_vector_type(2))) float v2f;
typedef __attribute__((ext_vector_type(8))) float v8f;

// ---------------------------------------------------------------------------
// Embed: f[b,n,o] = relu(bn(w@x + b)), also transpose xyz to (B,N,3)
// ---------------------------------------------------------------------------
__global__ void em_embed(const float* __restrict__ x, const float* __restrict__ w,
                         const float* __restrict__ bias,
                         const float* __restrict__ bg, const float* __restrict__ bb,
                         const float* __restrict__ bm, const float* __restrict__ bv,
                         float* __restrict__ f, float* __restrict__ xyz, int N) {
  int i = blockIdx.x * blockDim.x + threadIdx.x;
  if (i >= 16 * N) return;
  int b = i / N, n = i % N;
  float x0 = x[((long)b * 3 + 0) * N + n];
  float x1 = x[((long)b * 3 + 1) * N + n];
  float x2 = x[((long)b * 3 + 2) * N + n];
  xyz[((long)b * N + n) * 3 + 0] = x0;
  xyz[((long)b * N + n) * 3 + 1] = x1;
  xyz[((long)b * N + n) * 3 + 2] = x2;
  for (int o = 0; o < 24; ++o) {
    float h = w[o * 3 + 0] * x0 + w[o * 3 + 1] * x1 + w[o * 3 + 2] * x2 + bias[o];
    h = (h - bm[o]) * rsqrtf(bv[o] + 1e-5f) * bg[o] + bb[o];
    f[((long)b * N + n) * 24 + o] = fmaxf(h, 0.f);
  }
}

// ---------------------------------------------------------------------------
// Farthest point sampling: one block per batch, sequential over S.
// ---------------------------------------------------------------------------
__global__ void em_fps(const float* __restrict__ xyz, int* __restrict__ idx,
                       float* __restrict__ dists, int N, int S) {
  __shared__ float rv[256];
  __shared__ int   ri[256];
  __shared__ int   sfar;
  int b = blockIdx.x, t = threadIdx.x;
  float* dl = dists + (long)b * N;
  for (int n = t; n < N; n += 256) dl[n] = 1e10f;
  if (t == 0) sfar = 0;
  __syncthreads();
  for (int s = 0; s < S; ++s) {
    int far = sfar;
    if (t == 0) idx[b * S + s] = far;
    float cx = xyz[((long)b * N + far) * 3 + 0];
    float cy = xyz[((long)b * N + far) * 3 + 1];
    float cz = xyz[((long)b * N + far) * 3 + 2];
    float bv = -1e30f; int bi = 0x7fffffff;
    for (int n = t; n < N; n += 256) {
      float dx = xyz[((long)b * N + n) * 3 + 0] - cx;
      float dy = xyz[((long)b * N + n) * 3 + 1] - cy;
      float dz = xyz[((long)b * N + n) * 3 + 2] - cz;
      float d = dx * dx + dy * dy + dz * dz;
      float m = fminf(dl[n], d);
      dl[n] = m;
      if (m > bv) { bv = m; bi = n; }
    }
    rv[t] = bv; ri[t] = bi;
    __syncthreads();
    for (int o = 128; o > 0; o >>= 1) {
      if (t < o) {
        if (rv[t + o] > rv[t] || (rv[t + o] == rv[t] && ri[t + o] < ri[t])) {
          rv[t] = rv[t + o]; ri[t] = ri[t + o];
        }
      }
      __syncthreads();
    }
    if (t == 0) sfar = ri[0];
    __syncthreads();
  }
}

// ---------------------------------------------------------------------------
// Gather new_xyz, new_pts at FPS indices
// ---------------------------------------------------------------------------
__global__ void em_gather(const float* __restrict__ xyz, const float* __restrict__ f,
                          const int* __restrict__ idx, float* __restrict__ nxyz,
                          float* __restrict__ npts, int N, int S, int C) {
  int i = blockIdx.x * blockDim.x + threadIdx.x;
  if (i >= 16 * S) return;
  int b = i / S, s = i % S;
  int id = idx[b * S + s];
  for (int c = 0; c < 3; ++c) nxyz[(long)i * 3 + c] = xyz[((long)b * N + id) * 3 + c];
  for (int c = 0; c < C; ++c) npts[(long)i * C + c] = f[((long)b * N + id) * C + c];
}

// ---------------------------------------------------------------------------
// KNN: one block per (b,s); 24 argmin passes over d2 row in LDS.
// ---------------------------------------------------------------------------
__global__ void em_knn(const float* __restrict__ xyz, const float* __restrict__ nxyz,
                       int* __restrict__ knn, int N, int S) {
  __shared__ float sd[2048];
  __shared__ float rv[128];
  __shared__ int   ri[128];
  int gs = blockIdx.x, t = threadIdx.x;
  int b = gs / S;
  float qx = nxyz[(long)gs * 3 + 0], qy = nxyz[(long)gs * 3 + 1], qz = nxyz[(long)gs * 3 + 2];
  float qq = qx * qx + qy * qy + qz * qz;
  for (int n = t; n < N; n += 128) {
    float px = xyz[((long)b * N + n) * 3 + 0];
    float py = xyz[((long)b * N + n) * 3 + 1];
    float pz = xyz[((long)b * N + n) * 3 + 2];
    float pp = px * px + py * py + pz * pz;
    sd[n] = qq + pp - 2.f * (qx * px + qy * py + qz * pz);
  }
  __syncthreads();
  for (int j = 0; j < 24; ++j) {
    float bv = 3.4e38f; int bi = 0x7fffffff;
    for (int n = t; n < N; n += 128) {
      float v = sd[n];
      if (v < bv) { bv = v; bi = n; }
    }
    rv[t] = bv; ri[t] = bi;
    __syncthreads();
    for (int o = 64; o > 0; o >>= 1) {
      if (t < o) {
        if (rv[t + o] < rv[t] || (rv[t + o] == rv[t] && ri[t + o] < ri[t])) {
          rv[t] = rv[t + o]; ri[t] = ri[t + o];
        }
      }
      __syncthreads();
    }
    if (t == 0) { knn[(long)gs * 24 + j] = ri[0]; sd[ri[0]] = 3.4e38f; }
    __syncthreads();
  }
}

__global__ void em_zero2(float* a, float* b) {
  int t = threadIdx.x;
  if (t < 16) { a[t] = 0.f; b[t] = 0.f; }
}

// Accumulate per-batch sum / sumsq of diff (blocks never straddle a batch:
// S*24*ci == 294912 for all stages, a multiple of 256)
__global__ void em_diff_stats(const float* __restrict__ f, const float* __restrict__ npts,
                              const int* __restrict__ knn, float* __restrict__ ssum,
                              float* __restrict__ ssq, int N, int S, int ci) {
  __shared__ float s1[256], s2[256];
  int t = threadIdx.x;
  long i = (long)blockIdx.x * 256 + t;
  long perb = (long)S * 24 * ci;
  long tot = 16 * perb;
  float d = 0.f, dd = 0.f;
  int b0 = (int)(((long)blockIdx.x * 256) / perb);
  if (i < tot) {
    int b = (int)(i / perb);
    long r = i % perb;
    int s = (int)(r / (24 * ci));
    long r2 = r % (24 * ci);
    int k = (int)(r2 / ci);
    int c = (int)(r2 % ci);
    int id = knn[((long)b * S + s) * 24 + k];
    float diff = f[((long)b * N + id) * ci + c] - npts[((long)b * S + s) * ci + c];
    d = diff; dd = diff * diff;
  }
  s1[t] = d; s2[t] = dd;
  __syncthreads();
  for (int o = 128; o > 0; o >>= 1) {
    if (t < o) { s1[t] += s1[t + o]; s2[t] += s2[t + o]; }
    __syncthreads();
  }
  if (t == 0) { atomicAdd(&ssum[b0], s1[0]); atomicAdd(&ssq[b0], s2[0]); }
}

__global__ void em_std(const float* ssum, const float* ssq, float* sstd, int M) {
  int t = threadIdx.x;
  if (t < 16) {
    float s = ssum[t], q = ssq[t];
    float var = (q - s * s / (float)M) / (float)(M - 1);
    sstd[t] = sqrtf(var);
  }
}

// Build grouped tensor G (group, 2ci, 24): [normalized diff ; new_pts]
__global__ void em_buildG(const float* __restrict__ f, const float* __restrict__ npts,
                          const int* __restrict__ knn, const float* __restrict__ sstd,
                          const float* __restrict__ alpha, const float* __restrict__ beta,
                          float* __restrict__ G, int N, int S, int ci) {
  long i = (long)blockIdx.x * 256 + threadIdx.x;
  long perb = (long)S * 24 * ci;
  if (i >= 16 * perb) return;
  int b = (int)(i / perb);
  long r = i % perb;
  int s = (int)(r / (24 * ci));
  long r2 = r % (24 * ci);
  int k = (int)(r2 / ci);
  int c = (int)(r2 % ci);
  long gidx = (long)b * S + s;
  int id = knn[gidx * 24 + k];
  float np = npts[gidx * ci + c];
  float diff = f[((long)b * N + id) * ci + c] - np;
  float g = diff / (sstd[b] + 1e-5f) * alpha[c] + beta[c];
  int co2 = 2 * ci;
  G[(gidx * co2 + c) * 24 + k] = g;
  G[(gidx * co2 + ci + c) * 24 + k] = np;
}

// ---------------------------------------------------------------------------
// Generic fp32 WMMA GEMM:  Y = act( bn( W @ X + bias ) + res )
// W: (M x Kd) row-major shared; X/Y/res strided per blockIdx.z.
// One wave per 16x16 output tile; K stepped by 4 via V_WMMA_F32_16X16X4_F32.
//
// Requirements: Kd % 4 == 0 (true for every call site here).
// M/N edges are handled WITHOUT masking loads: out-of-range rows/cols are
// CLAMPED (A row m only feeds D row m; B col n only feeds D col n; those
// positions are masked at the store), so the inner loop is branch-free with
// EXEC all-1s — unconditional b64-mergeable loads feeding v_wmma.
// ---------------------------------------------------------------------------
__global__ void em_gemm(const float* __restrict__ W, const float* __restrict__ X,
                        float* __restrict__ Y, const float* __restrict__ bias,
                        const float* __restrict__ bng, const float* __restrict__ bnb,
                        const float* __restrict__ bnm, const float* __restrict__ bnv,
                        const float* __restrict__ res, int M, int Kd, int N,
                        long xB, int xk, int xn, long yB, int yk, int yn, int relu) {
  int lane = threadIdx.x;
  int half = lane >> 4, q = lane & 15;
  int n0 = blockIdx.x * 16, m0 = blockIdx.y * 16;
  long gz = blockIdx.z;

  int arow = m0 + q; if (arow > M - 1) arow = M - 1;   // clamp, no zeroing needed
  int bcol = n0 + q; if (bcol > N - 1) bcol = N - 1;

  const float* Wr = W + (long)arow * Kd + 2 * half;             // row of A, K offset
  const float* Xc = X + gz * xB + (long)bcol * xn + (long)(2 * half) * xk;

  v8f acc = {0.f, 0.f, 0.f, 0.f, 0.f, 0.f, 0.f, 0.f};
  for (int k0 = 0; k0 < Kd; k0 += 4) {
    v2f a, bb;
    // A 16x4: lanes 0-15 hold K{0,1}, lanes 16-31 hold K{2,3}; contiguous pair
    a.x = Wr[0];
    a.y = Wr[1];
    // B 4x16: vgpr0 rows {0,2}, vgpr1 rows {1,3}; col = lane&15
    bb.x = Xc[0];
    bb.y = Xc[xk];
    acc = __builtin_amdgcn_wmma_f32_16x16x4_f32(false, a, false, bb, (short)0, acc,
                                                false, false);
    Wr += 4;
    Xc += (long)4 * xk;
  }

  float* Yb = Y + gz * yB;
  const float* Rb = res ? (res + gz * yB) : (const float*)0;
  for (int r = 0; r < 8; ++r) {
    int m = m0 + r + 8 * half;   // C/D: vgpr r -> M=r (lanes<16) / r+8 (lanes>=16)
    int n = n0 + q;
    if (m < M && n < N) {
      float v = acc[r];
      if (bias) v += bias[m];
      if (bng)  v = (v - bnm[m]) * rsqrtf(bnv[m] + 1e-5f) * bng[m] + bnb[m];
      if (Rb)   v += Rb[(long)m * yk + (long)n * yn];
      if (relu) v = fmaxf(v, 0.f);
      Yb[(long)m * yk + (long)n * yn] = v;
    }
  }
}

// ---------------------------------------------------------------------------
// Attention: logits over T tokens, softmax, weighted feature sum.
// ---------------------------------------------------------------------------
__global__ void em_att(const float* __restrict__ X, const float* __restrict__ w,
                       const float* __restrict__ bp, float* __restrict__ out,
                       int T, int D, long gX, int tX, int dX, long gO, int dO) {
  __shared__ float sl[128];
  __shared__ float sr[128];
  int g = blockIdx.x, t = threadIdx.x;
  const float* Xg = X + (long)g * gX;
  float lg = -3.4e38f;
  if (t < T) {
    float s = bp[0];
    for (int d = 0; d < D; ++d) s += Xg[(long)t * tX + (long)d * dX] * w[d];
    lg = s;
  }
  sl[t] = lg; sr[t] = lg;
  __syncthreads();
  for (int o = 64; o > 0; o >>= 1) {
    if (t < o) sr[t] = fmaxf(sr[t], sr[t + o]);
    __syncthreads();
  }
  float mx = sr[0];
  __syncthreads();
  float e = (t < T) ? __expf(sl[t] - mx) : 0.f;
  sl[t] = e; sr[t] = e;
  __syncthreads();
  for (int o = 64; o > 0; o >>= 1) {
    if (t < o) sr[t] += sr[t + o];
    __syncthreads();
  }
  float sm = sr[0];
  __syncthreads();
  sl[t] = sl[t] / sm;
  __syncthreads();
  for (int d = t; d < D; d += 128) {
    float s = 0.f;
    for (int k = 0; k < T; ++k) s += sl[k] * Xg[(long)k * tX + (long)d * dX];
    out[(long)g * gO + (long)d * dO] = s;
  }
}

__global__ void em_ln(const float* __restrict__ X, const float* __restrict__ w,
                      const float* __restrict__ b, float* __restrict__ Y, int D) {
  __shared__ float s1[64], s2[64];
  long g = blockIdx.x;
  int t = threadIdx.x;
  const float* xg = X + g * D;
  float a = 0.f, aa = 0.f;
  for (int d = t; d < D; d += 64) { float v = xg[d]; a += v; aa += v * v; }
  s1[t] = a; s2[t] = aa;
  __syncthreads();
  for (int o = 32; o > 0; o >>= 1) {
    if (t < o) { s1[t] += s1[t + o]; s2[t] += s2[t + o]; }
    __syncthreads();
  }
  float mu = s1[0] / D;
  float var = s2[0] / D - mu * mu;
  float rs = rsqrtf(var + 1e-5f);
  for (int d = t; d < D; d += 64) Y[g * D + d] = (xg[d] - mu) * rs * w[d] + b[d];
}

// Depthwise causal conv (K=4) + SiLU; rev handles the bwd branch's reversal.
__global__ void em_conv(const float* __restrict__ xz, const float* __restrict__ cw,
                        const float* __restrict__ cb, float* __restrict__ uc,
                        int di, int L, int rev) {
  long i = (long)blockIdx.x * 256 + threadIdx.x;
  if (i >= (long)16 * di * L) return;
  int b = (int)(i / ((long)di * L));
  long r = i % ((long)di * L);
  int d = (int)(r / L), l = (int)(r % L);
  const float* u = xz + ((long)b * 2 * di + d) * L;
  float acc = cb[d];
  for (int j = 0; j < 4; ++j) {
    int m = l - 3 + j;
    if (m >= 0) {
      int p = rev ? (L - 1 - m) : m;
      acc += cw[d * 4 + j] * u[p];
    }
  }
  uc[((long)b * di + d) * L + l] = acc / (1.f + __expf(-acc));
}

// delta = softplus(dt_proj_w @ dt + b)
__global__ void em_delta(const float* __restrict__ xdbl, const float* __restrict__ dtw,
                         const float* __restrict__ dtb, float* __restrict__ delta,
                         int di, int L, int dtr, int E) {
  long i = (long)blockIdx.x * 256 + threadIdx.x;
  if (i >= (long)16 * di * L) return;
  int b = (int)(i / ((long)di * L));
  long r = i % ((long)di * L);
  int d = (int)(r / L), l = (int)(r % L);
  const float* xd = xdbl + (long)b * E * L;
  float a = dtb[d];
  for (int rr = 0; rr < dtr; ++rr) a += dtw[d * dtr + rr] * xd[(long)rr * L + l];
  delta[((long)b * di + d) * L + l] = fmaxf(a, 0.f) + log1pf(__expf(-fabsf(a)));
}

// Selective scan: one thread per (b,d), 16-state recurrence; SiLU(z)-gated out.
__global__ void em_scan(const float* __restrict__ uc, const float* __restrict__ delta,
                        const float* __restrict__ xdbl, const float* __restrict__ Alog,
                        const float* __restrict__ Dp, const float* __restrict__ xz,
                        float* __restrict__ y, int di, int L, int dtr, int E, int rev) {
  int i = blockIdx.x * 128 + threadIdx.x;
  if (i >= 16 * di) return;
  int b = i / di, d = i % di;
  float An[16];
  for (int n = 0; n < 16; ++n) An[n] = -__expf(Alog[d * 16 + n]);
  float h[16];
  for (int n = 0; n < 16; ++n) h[n] = 0.f;
  float Dd = Dp[d];
  const float* BM = xdbl + (long)b * E * L + (long)dtr * L;
  const float* CM = BM + (long)16 * L;
  long base = ((long)b * di + d) * L;
  const float* zrow = xz + ((long)b * 2 * di + di + d) * L;
  for (int l = 0; l < L; ++l) {
    float dl = delta[base + l];
    float u = uc[base + l];
    float yv = 0.f;
    for (int n = 0; n < 16; ++n) {
      float hn = __expf(dl * An[n]) * h[n] + dl * BM[(long)n * L + l] * u;
      h[n] = hn;
      yv += hn * CM[(long)n * L + l];
    }
    yv += u * Dd;
    int zl = rev ? (L - 1 - l) : l;
    float z = zrow[zl];
    y[base + l] = yv * (z / (1.f + __expf(-z)));
  }
}

__global__ void em_combine(const float* __restrict__ yf, const float* __restrict__ yb,
                           float* __restrict__ ys, int di, int L) {
  long i = (long)blockIdx.x * 256 + threadIdx.x;
  if (i >= (long)16 * di * L) return;
  long row = i / L;
  int l = (int)(i % L);
  ys[row * L + l] = yf[row * L + l] + yb[row * L + (L - 1 - l)];
}

__global__ void em_transpose(const float* __restrict__ Fc, float* __restrict__ f,
                             int C, int L) {
  long i = (long)blockIdx.x * 256 + threadIdx.x;
  if (i >= (long)16 * C * L) return;
  int b = (int)(i / ((long)C * L));
  long r = i % ((long)C * L);
  int c = (int)(r / L), l = (int)(r % L);
  f[((long)b * L + l) * C + c] = Fc[((long)b * C + c) * L + l];
}

__global__ void em_cls2(const float* __restrict__ h, const float* __restrict__ w2,
                        const float* __restrict__ b2, float* __restrict__ out) {
  int t = threadIdx.x;
  if (t >= 160) return;
  int b = t / 10, o = t % 10;
  float s = b2[o];
  for (int c = 0; c < 256; ++c) s += w2[o * 256 + c] * h[c * 16 + b];
  out[b * 10 + o] = s;
}

// ---------------------------------------------------------------------------
// Host orchestration
// ---------------------------------------------------------------------------
extern "C" void kernel_launch(void* const* d_in, const int* in_sizes, int n_in,
                              void* d_out, int out_size, void* d_ws, size_t ws_size,
                              hipStream_t stream) {
  (void)in_sizes; (void)n_in; (void)out_size; (void)ws_size;
  int pi = 0;
  auto nx = [&]() { return (const float*)d_in[pi++]; };

  struct BNp { const float *g, *b, *m, *v; };
  struct L2p { const float *w1, *b1; BNp bn1; const float *w2, *b2; BNp bn2; };
  struct Brp { const float *cw, *cb, *xw, *dtw, *dtb, *al, *D; };
  struct Stp {
    const float *alpha, *beta;
    L2p loc;
    const float *aw, *ab, *nw, *nb, *inw, *outw;
    Brp fw, bw;
    L2p gl;
  };

  auto getBN = [&]() { BNp o; o.g = nx(); o.b = nx(); o.m = nx(); o.v = nx(); return o; };
  auto getL2 = [&]() { L2p o; o.w1 = nx(); o.b1 = nx(); o.bn1 = getBN();
                       o.w2 = nx(); o.b2 = nx(); o.bn2 = getBN(); return o; };
  auto getBr = [&]() { Brp o; o.cw = nx(); o.cb = nx(); o.xw = nx(); o.dtw = nx();
                       o.dtb = nx(); o.al = nx(); o.D = nx(); return o; };

  const float* x = nx();
  const float* ew = nx(); const float* ebias = nx();
  const float *ebg = nx(), *ebb = nx(), *ebm = nx(), *ebv = nx();
  Stp st[3];
  for (int i = 0; i < 3; ++i) {
    st[i].alpha = nx(); st[i].beta = nx();
    st[i].loc = getL2();
    st[i].aw = nx(); st[i].ab = nx();
    st[i].nw = nx(); st[i].nb = nx();
    st[i].inw = nx(); st[i].outw = nx();
    st[i].fw = getBr(); st[i].bw = getBr();
    st[i].gl = getL2();
  }
  const float* attw = nx(); const float* attb = nx();
  const float* cw1 = nx(); const float* cb1 = nx();
  BNp cbn = getBN();
  const float* cw2 = nx(); const float* cb2 = nx();

  // workspace bump allocator
  float* wp = (float*)d_ws;
  auto allocF = [&](long n) { float* r = wp; wp += ((n + 63) & ~63L); return r; };
  float* xyzA = allocF(98304);
  float* xyzB = allocF(98304);
  float* fA   = allocF(786432);
  float* fB   = allocF(786432);
  float* dists = allocF(32768);
  int*   idxb = (int*)allocF(8192);
  float* npts = allocF(196608);
  int*   knn  = (int*)allocF(196608);
  float* ssum = allocF(64);
  float* ssq  = allocF(64);
  float* sstd = allocF(64);
  float* G    = allocF(9437184);
  float* H1   = allocF(4718592);
  float* fbn  = allocF(393216);
  float* hln  = allocF(393216);
  float* xz   = allocF(1572864);
  float* ucF  = allocF(786432);
  float* ucB  = allocF(786432);
  float* xdF  = allocF(393216);
  float* xdB  = allocF(393216);
  float* deF  = allocF(786432);
  float* deB  = allocF(786432);
  float* yF   = allocF(786432);
  float* yB   = allocF(786432);
  float* ysum = allocF(786432);
  float* Fc   = allocF(393216);
  float* T1   = allocF(196608);
  float* vT   = allocF(3072);
  float* hcls = allocF(4096);

  auto gemm = [&](const float* W, const float* X, float* Y, const float* bias,
                  const BNp* bn, const float* res, int M, int K, int N,
                  long xB, int xk, int xn, long yB, int yk, int yn, int nb, int relu) {
    dim3 g(CDIV(N, 16), CDIV(M, 16), nb);
    em_gemm<<<g, 32, 0, stream>>>(W, X, Y, bias,
        bn ? bn->g : nullptr, bn ? bn->b : nullptr,
        bn ? bn->m : nullptr, bn ? bn->v : nullptr,
        res, M, K, N, xB, xk, xn, yB, yk, yn, relu);
  };

  const int FEAT[4] = {24, 48, 96, 192};
  const int GR[3] = {512, 256, 128};

  float* f_cur = fA; float* f_nxt = fB;
  float* xyz_cur = xyzA; float* xyz_nxt = xyzB;
  int Np = 2048;

  em_embed<<<CDIV(16 * Np, 256), 256, 0, stream>>>(x, ew, ebias, ebg, ebb, ebm, ebv,
                                                   f_cur, xyz_cur, Np);

  for (int i = 0; i < 3; ++i) {
    int ci = FEAT[i], co = FEAT[i + 1], S = GR[i], L = S;
    int di = 2 * co, dtr = (co + 15) / 16, E = dtr + 32;
    Stp& P = st[i];

    em_fps<<<16, 256, 0, stream>>>(xyz_cur, idxb, dists, Np, S);
    em_gather<<<CDIV(16 * S, 256), 256, 0, stream>>>(xyz_cur, f_cur, idxb,
                                                     xyz_nxt, npts, Np, S, ci);
    em_knn<<<16 * S, 128, 0, stream>>>(xyz_cur, xyz_nxt, knn, Np, S);
    em_zero2<<<1, 16, 0, stream>>>(ssum, ssq);
    long tot = (long)16 * S * 24 * ci;
    em_diff_stats<<<CDIV(tot, 256), 256, 0, stream>>>(f_cur, npts, knn, ssum, ssq,
                                                      Np, S, ci);
    em_std<<<1, 16, 0, stream>>>(ssum, ssq, sstd, S * 24 * ci);
    em_buildG<<<CDIV(tot, 256), 256, 0, stream>>>(f_cur, npts, knn, sstd,
                                                  P.alpha, P.beta, G, Np, S, ci);
    int nb = 16 * S;
    // local lin2: per-group GEMMs (N = 24 neighbors)
    gemm(P.loc.w1, G, H1, P.loc.b1, &P.loc.bn1, nullptr, co / 2, co, 24,
         (long)co * 24, 24, 1, (long)(co / 2) * 24, 24, 1, nb, 1);
    gemm(P.loc.w2, H1, G, P.loc.b2, &P.loc.bn2, G, co, co / 2, 24,
         (long)(co / 2) * 24, 24, 1, (long)co * 24, 24, 1, nb, 1);
    // attention over K=24 neighbors -> (B,S,co)
    em_att<<<nb, 128, 0, stream>>>(G, P.aw, P.ab, fbn, 24, co,
                                   (long)co * 24, 1, 24, (long)co, 1);
    em_ln<<<nb, 64, 0, stream>>>(fbn, P.nw, P.nb, hln, co);
    // mamba in_proj: (2di x co) @ (co x L) per batch
    gemm(P.inw, hln, xz, nullptr, nullptr, nullptr, 2 * di, co, L,
         (long)L * co, 1, co, (long)2 * di * L, L, 1, 16, 0);
    long ndl = (long)16 * di * L;
    em_conv<<<CDIV(ndl, 256), 256, 0, stream>>>(xz, P.fw.cw, P.fw.cb, ucF, di, L, 0);
    em_conv<<<CDIV(ndl, 256), 256, 0, stream>>>(xz, P.bw.cw, P.bw.cb, ucB, di, L, 1);
    gemm(P.fw.xw, ucF, xdF, nullptr, nullptr, nullptr, E, di, L,
         (long)di * L, L, 1, (long)E * L, L, 1, 16, 0);
    gemm(P.bw.xw, ucB, xdB, nullptr, nullptr, nullptr, E, di, L,
         (long)di * L, L, 1, (long)E * L, L, 1, 16, 0);
    em_delta<<<CDIV(ndl, 256), 256, 0, stream>>>(xdF, P.fw.dtw, P.fw.dtb, deF,
                                                 di, L, dtr, E);
    em_delta<<<CDIV(ndl, 256), 256, 0, stream>>>(xdB, P.bw.dtw, P.bw.dtb, deB,
                                                 di, L, dtr, E);
    em_scan<<<CDIV(16 * di, 128), 128, 0, stream>>>(ucF, deF, xdF, P.fw.al, P.fw.D,
                                                    xz, yF, di, L, dtr, E, 0);
    em_scan<<<CDIV(16 * di, 128), 128, 0, stream>>>(ucB, deB, xdB, P.bw.al, P.bw.D,
                                                    xz, yB, di, L, dtr, E, 1);
    em_combine<<<CDIV(ndl, 256), 256, 0, stream>>>(yF, yB, ysum, di, L);
    gemm(P.outw, ysum, Fc, nullptr, nullptr, nullptr, co, di, L,
         (long)di * L, L, 1, (long)co * L, L, 1, 16, 0);
    // global lin2 on (B, co, L)
    gemm(P.gl.w1, Fc, T1, P.gl.b1, &P.gl.bn1, nullptr, co / 2, co, L,
         (long)co * L, L, 1, (long)(co / 2) * L, L, 1, 16, 1);
    gemm(P.gl.w2, T1, Fc, P.gl.b2, &P.gl.bn2, Fc, co, co / 2, L,
         (long)(co / 2) * L, L, 1, (long)co * L, L, 1, 16, 1);
    em_transpose<<<CDIV((long)16 * co * L, 256), 256, 0, stream>>>(Fc, f_nxt, co, L);

    { float* t0 = f_cur; f_cur = f_nxt; f_nxt = t0; }
    { float* t1 = xyz_cur; xyz_cur = xyz_nxt; xyz_nxt = t1; }
    Np = S;
  }

  // final attention over 128 tokens of dim 192 -> vT (192 x 16)
  em_att<<<16, 128, 0, stream>>>(f_cur, attw, attb, vT, 128, 192,
                                 (long)128 * 192, 192, 1, 1L, 16);
  // classifier layer 1: (256 x 192) @ (192 x 16) with BN+ReLU
  gemm(cw1, vT, hcls, cb1, &cbn, nullptr, 256, 192, 16,
       0L, 16, 1, 0L, 16, 1, 1, 1);
  em_cls2<<<1, 160, 0, stream>>>(hcls, cw2, cb2, (float*)d_out);
}